// opt_kde_53987738911382
// MI455X (gfx1250) — compile-verified
//
#include <hip/hip_runtime.h>
#include <math.h>

#define EE 4
#define LL 5
#define NN 512
#define FF 128
#define SS 1000

typedef float v2f __attribute__((ext_vector_type(2)));
typedef float v8f __attribute__((ext_vector_type(8)));

// ---------------------------------------------------------------------------
// Kernel 1: proj[e,l,n,g] = sum_f matrix[e,l,n,f] * params[f,g]
// Flattened GEMM: [10240,128] x [128,128], fp32, via V_WMMA_F32_16X16X4_F32.
// One wave per 16x16 output tile; 8 waves (256 threads) per block.
// ---------------------------------------------------------------------------
__global__ __launch_bounds__(256) void proj_wmma_kernel(
    const float* __restrict__ A, const float* __restrict__ B,
    float* __restrict__ C) {
  const int wave = blockIdx.x * 8 + (threadIdx.x >> 5);
  const int lane = threadIdx.x & 31;
  const int tm = wave >> 3;       // 0..639 (row tiles of 16)
  const int tn = wave & 7;        // 0..7   (col tiles of 16)
  const int rowBase = tm * 16;
  const int colBase = tn * 16;
  const int half = lane >> 4;     // 0 or 1
  const int l16  = lane & 15;

  const float* arow = A + (size_t)(rowBase + l16) * FF;
  v8f acc = {};
#pragma unroll 4
  for (int k0 = 0; k0 < FF; k0 += 4) {
    const int ka = k0 + 2 * half;
    v2f a, b;
    a.x = arow[ka];
    a.y = arow[ka + 1];
    b.x = B[(size_t)ka * FF + colBase + l16];
    b.y = B[(size_t)(ka + 1) * FF + colBase + l16];
    acc = __builtin_amdgcn_wmma_f32_16x16x4_f32(
        /*neg_a=*/false, a, /*neg_b=*/false, b,
        /*c_mod=*/(short)0, acc, /*reuse_a=*/false, /*reuse_b=*/false);
  }
#pragma unroll
  for (int i = 0; i < 8; ++i) {
    C[(size_t)(rowBase + i + 8 * half) * FF + colBase + l16] = acc[i];
  }
}

// ---------------------------------------------------------------------------
// Kernel 2: per-(e,l) partials: sum of std over F columns (ddof=1, padded
// data included, matching reference), plus min/max of proj.
// 20 blocks x 128 threads (thread = column f).
// ---------------------------------------------------------------------------
__global__ __launch_bounds__(128) void stats_part_kernel(
    const float* __restrict__ mat, const float* __restrict__ proj,
    float* __restrict__ partials) {
  const int b = blockIdx.x;     // e*LL + l
  const int f = threadIdx.x;    // 0..127
  const float* mcol = mat + (size_t)b * NN * FF + f;
  float sum = 0.f, sq = 0.f;
  for (int n = 0; n < NN; ++n) {
    float v = mcol[(size_t)n * FF];
    sum += v; sq += v * v;
  }
  float var = (sq - sum * sum / (float)NN) / (float)(NN - 1);
  float sd = sqrtf(fmaxf(var, 0.f));

  const float* pcol = proj + (size_t)b * NN * FF + f;
  float mn = 3.4e38f, mx = -3.4e38f;
  for (int n = 0; n < NN; ++n) {
    float v = pcol[(size_t)n * FF];
    mn = fminf(mn, v); mx = fmaxf(mx, v);
  }

  __shared__ float r0[128], r1[128], r2[128];
  r0[f] = sd; r1[f] = mn; r2[f] = mx;
  __syncthreads();
  for (int off = 64; off > 0; off >>= 1) {
    if (f < off) {
      r0[f] += r0[f + off];
      r1[f] = fminf(r1[f], r1[f + off]);
      r2[f] = fmaxf(r2[f], r2[f + off]);
    }
    __syncthreads();
  }
  if (f == 0) {
    partials[b * 3 + 0] = r0[0];
    partials[b * 3 + 1] = r1[0];
    partials[b * 3 + 2] = r2[0];
  }
}

// ---------------------------------------------------------------------------
// Kernel 3: finalize scalars {w = c*log2(e), left, step, delta, divisor}.
// ---------------------------------------------------------------------------
__global__ void stats_final_kernel(const float* __restrict__ partials,
                                   float* __restrict__ sc) {
  if (threadIdx.x == 0) {
    float ssum = 0.f, mn = 3.4e38f, mx = -3.4e38f;
    for (int b = 0; b < EE * LL; ++b) {
      ssum += partials[b * 3 + 0];
      mn = fminf(mn, partials[b * 3 + 1]);
      mx = fmaxf(mx, partials[b * 3 + 2]);
    }
    float meanstd = ssum / (float)(EE * LL * FF);
    float bw = 1.06f * powf((float)NN, -0.2f) * meanstd;
    float c = -0.5f / (bw * bw);
    float divisor = sqrtf(2.0f * (float)M_PI) * bw;
    sc[0] = c * 1.44269504088896340736f;  // w: base-2 exponent scale
    sc[1] = mn;                           // left
    sc[2] = (mx - mn) / (float)(SS - 1);  // linspace step
    sc[3] = (mx - mn) / (float)SS;        // delta
    sc[4] = divisor;
  }
}

// ---------------------------------------------------------------------------
// Kernel 4: heavy KDE scan. One block per (l,f) pair (640 blocks).
// exp(c(x-p)^2) = exp2(w p^2) * exp2(fma(-2wx, p, wx^2)); the u = w x^2 and
// v = -2wx tables live in LDS, the exp2(w p^2) factor is hoisted per grid
// point and folded into the padding correction. Inner body per sample:
// v_fma_f32 + v_exp_f32 + v_add_f32. Only the 6 unique env pairs are
// accumulated (|.| is symmetric, diagonal is zero).
// ---------------------------------------------------------------------------
__global__ __launch_bounds__(256) void kde_kernel(
    const float* __restrict__ proj, const int* __restrict__ dlen,
    const float* __restrict__ sc, float* __restrict__ storeSum) {
  const int blk = blockIdx.x;     // l*FF + f
  const int l = blk / FF;
  const int f = blk % FF;
  const int tid = threadIdx.x;

  __shared__ float4 us[NN];       // 8 KB: w*x^2 per env
  __shared__ float4 vs[NN];       // 8 KB: -2*w*x per env
  __shared__ float red[256];

  const float w = sc[0];
  const float left = sc[1];
  const float step = sc[2];

  for (int i = tid; i < NN; i += 256) {
    float x0 = proj[((size_t)((0 * LL + l) * NN + i)) * FF + f];
    float x1 = proj[((size_t)((1 * LL + l) * NN + i)) * FF + f];
    float x2 = proj[((size_t)((2 * LL + l) * NN + i)) * FF + f];
    float x3 = proj[((size_t)((3 * LL + l) * NN + i)) * FF + f];
    float4 u, v;
    u.x = w * x0 * x0;  v.x = -2.f * w * x0;
    u.y = w * x1 * x1;  v.y = -2.f * w * x1;
    u.z = w * x2 * x2;  v.z = -2.f * w * x2;
    u.w = w * x3 * x3;  v.w = -2.f * w * x3;
    us[i] = u; vs[i] = v;
  }
  const float len0 = (float)dlen[0 * LL + l];
  const float len1 = (float)dlen[1 * LL + l];
  const float len2 = (float)dlen[2 * LL + l];
  const float len3 = (float)dlen[3 * LL + l];
  const float il0 = 1.f / len0, il1 = 1.f / len1;
  const float il2 = 1.f / len2, il3 = 1.f / len3;
  const float of0 = ((float)NN - len0) * il0;
  const float of1 = ((float)NN - len1) * il1;
  const float of2 = ((float)NN - len2) * il2;
  const float of3 = ((float)NN - len3) * il3;
  __syncthreads();

  float a01 = 0.f, a02 = 0.f, a03 = 0.f, a12 = 0.f, a13 = 0.f, a23 = 0.f;

  for (int s = tid; s < SS; s += 256) {
    const float p = left + step * (float)s;
    float s0 = 0.f, s1 = 0.f, s2 = 0.f, s3 = 0.f;
#pragma unroll 4
    for (int n = 0; n < NN; ++n) {
      float4 u = us[n];
      float4 v = vs[n];
      s0 += __builtin_amdgcn_exp2f(fmaf(v.x, p, u.x));
      s1 += __builtin_amdgcn_exp2f(fmaf(v.y, p, u.y));
      s2 += __builtin_amdgcn_exp2f(fmaf(v.z, p, u.z));
      s3 += __builtin_amdgcn_exp2f(fmaf(v.w, p, u.w));
    }
    const float E2 = __builtin_amdgcn_exp2f(w * p * p);  // == exp(c p^2)
    const float k0 = E2 * (s0 * il0 - of0);
    const float k1 = E2 * (s1 * il1 - of1);
    const float k2 = E2 * (s2 * il2 - of2);
    const float k3 = E2 * (s3 * il3 - of3);
    a01 += fabsf(k0 - k1);
    a02 += fabsf(k0 - k2);
    a03 += fabsf(k0 - k3);
    a12 += fabsf(k1 - k2);
    a13 += fabsf(k1 - k3);
    a23 += fabsf(k2 - k3);
  }

  float accq[6] = {a01, a02, a03, a12, a13, a23};
  const int pe1[6] = {0, 0, 0, 1, 1, 2};
  const int pe2[6] = {1, 2, 3, 2, 3, 3};
  for (int q = 0; q < 6; ++q) {
    red[tid] = accq[q];
    __syncthreads();
    for (int off = 128; off > 0; off >>= 1) {
      if (tid < off) red[tid] += red[tid + off];
      __syncthreads();
    }
    if (tid == 0) {
      float r = red[0];
      storeSum[(size_t)(pe1[q] * EE + pe2[q]) * (LL * FF) + blk] = r;
      storeSum[(size_t)(pe2[q] * EE + pe1[q]) * (LL * FF) + blk] = r;
    }
    __syncthreads();
  }
  if (tid == 0) {
#pragma unroll
    for (int e = 0; e < EE; ++e)
      storeSum[(size_t)(e * EE + e) * (LL * FF) + blk] = 0.f;
  }
}

// ---------------------------------------------------------------------------
// Kernel 5: scale, pairwise maxes, outputs.
// out layout: train_results[5*128] | test_results[5*128] | train_dis | test_dis
// ---------------------------------------------------------------------------
__global__ __launch_bounds__(128) void final_kernel(
    const float* __restrict__ storeSum, const float* __restrict__ sc,
    float* __restrict__ out) {
  const int f = threadIdx.x;
  const float scale = sc[3] * 0.5f / sc[4];  // delta/2 / divisor
  const int trainPairs[9] = {0, 1, 2, 4, 5, 6, 8, 9, 10};

  float trmaxL = -3.4e38f, temaxL = -3.4e38f;
  for (int l = 0; l < LL; ++l) {
    float te = -3.4e38f, tr = -3.4e38f;
    for (int pr = 0; pr < 16; ++pr) {
      float v = storeSum[(size_t)pr * (LL * FF) + l * FF + f] * scale;
      te = fmaxf(te, v);
    }
#pragma unroll
    for (int i = 0; i < 9; ++i) {
      float v = storeSum[(size_t)trainPairs[i] * (LL * FF) + l * FF + f] * scale;
      tr = fmaxf(tr, v);
    }
    out[l * FF + f] = tr;                // train_results
    out[LL * FF + l * FF + f] = te;      // test_results
    trmaxL = fmaxf(trmaxL, tr);
    temaxL = fmaxf(temaxL, te);
  }

  __shared__ float ra[128], rb[128];
  ra[f] = trmaxL; rb[f] = temaxL;
  __syncthreads();
  for (int off = 64; off > 0; off >>= 1) {
    if (f < off) { ra[f] += ra[f + off]; rb[f] += rb[f + off]; }
    __syncthreads();
  }
  if (f == 0) {
    out[2 * LL * FF + 0] = ra[0] / (float)FF;  // train_dis
    out[2 * LL * FF + 1] = rb[0] / (float)FF;  // test_dis
  }
}

// ---------------------------------------------------------------------------
extern "C" void kernel_launch(void* const* d_in, const int* in_sizes, int n_in,
                              void* d_out, int out_size, void* d_ws,
                              size_t ws_size, hipStream_t stream) {
  (void)in_sizes; (void)n_in; (void)out_size; (void)ws_size;
  const float* matrix = (const float*)d_in[0];   // [4,5,512,128]
  const float* params = (const float*)d_in[1];   // [128,128]
  const int*   dlen   = (const int*)d_in[2];     // [4,5]

  float* ws       = (float*)d_ws;
  float* proj     = ws;                                   // 1,310,720 floats
  float* partials = ws + (size_t)EE * LL * NN * FF;       // 60 floats
  float* sc       = partials + 64;                        // 8 floats
  float* storeSum = sc + 8;                               // 10,240 floats

  proj_wmma_kernel<<<640, 256, 0, stream>>>(matrix, params, proj);
  stats_part_kernel<<<EE * LL, 128, 0, stream>>>(matrix, proj, partials);
  stats_final_kernel<<<1, 32, 0, stream>>>(partials, sc);
  kde_kernel<<<LL * FF, 256, 0, stream>>>(proj, dlen, sc, storeSum);
  final_kernel<<<1, 128, 0, stream>>>(storeSum, sc, (float*)d_out);
}